// FreqAdaptiveDecomp_3246995276119
// MI455X (gfx1250) — compile-verified
//
#include <hip/hip_runtime.h>
#include <math.h>

// Problem constants (from reference)
#define SEQ_LEN   4096
#define CHANNELS  512
#define F_BINS    2049          // SEQ_LEN/2 + 1
#define BATCH     16
#define LOG2N     12
#define PMAX      1024          // (SEQ_LEN/2 + 1)/2
#define XPLEN     (SEQ_LEN + 2*PMAX)   // 6144
#define NTHREADS  256
#define CHUNK     (XPLEN / NTHREADS)   // 24

typedef float v2f __attribute__((ext_vector_type(2)));
typedef float v8f __attribute__((ext_vector_type(8)));

__device__ __forceinline__ int brev12(int t) { return (int)(__brev((unsigned)t) >> 20); }

#define TWO_PI_F 6.28318530717958647692f

// ---------------------------------------------------------------------------
// CDNA5 async global->LDS copy (ASYNCcnt-tracked). LDS destination offset is
// the low 32 bits of the generic pointer (flat LDS aperture mapping).
// ---------------------------------------------------------------------------
__device__ __forceinline__ void async_load_f32_to_lds(const float* gptr, float* lptr) {
  unsigned lds_off = (unsigned)(uintptr_t)lptr;
  unsigned long long gaddr = (unsigned long long)(uintptr_t)gptr;
  asm volatile("global_load_async_to_lds_b32 %0, %1, off"
               :: "v"(lds_off), "v"(gaddr)
               : "memory");
}
__device__ __forceinline__ void wait_asynccnt0() {
  asm volatile("s_wait_asynccnt 0x0" ::: "memory");
}

// In-LDS radix-2 DIT FFT, length 4096. Input must be pre-permuted to
// bit-reversed order; output is natural order. sgn=-1 => forward transform.
__device__ void fft4096_dit(float* re, float* im, float sgn) {
  for (int s = 1; s <= LOG2N; ++s) {
    const int half = 1 << (s - 1);
    for (int j = threadIdx.x; j < (SEQ_LEN / 2); j += blockDim.x) {
      const int pos = j & (half - 1);
      const int i0  = ((j >> (s - 1)) << s) + pos;
      const int i1  = i0 + half;
      const float ang = sgn * TWO_PI_F * (float)pos / (float)(1 << s);
      float wr, wi;
      __sincosf(ang, &wi, &wr);              // w = cos + i*sin
      const float ur = re[i0], ui = im[i0];
      const float xr = re[i1], xi = im[i1];
      const float vr = xr * wr - xi * wi;
      const float vi = xr * wi + xi * wr;
      re[i0] = ur + vr;  im[i0] = ui + vi;
      re[i1] = ur - vr;  im[i1] = ui - vi;
    }
    __syncthreads();
  }
}

// In-LDS radix-2 DIF inverse FFT (twiddle sign +1). Input natural order,
// output BIT-REVERSED order (caller un-reverses at the final store).
__device__ void fft4096_dif_inv(float* re, float* im) {
  for (int s = LOG2N; s >= 1; --s) {
    const int half = 1 << (s - 1);
    for (int j = threadIdx.x; j < (SEQ_LEN / 2); j += blockDim.x) {
      const int pos = j & (half - 1);
      const int i0  = ((j >> (s - 1)) << s) + pos;
      const int i1  = i0 + half;
      const float ang = TWO_PI_F * (float)pos / (float)(1 << s);  // +i => inverse
      float wr, wi;
      __sincosf(ang, &wi, &wr);
      const float ur = re[i0], ui = im[i0];
      const float vr = re[i1], vi = im[i1];
      re[i0] = ur + vr;  im[i0] = ui + vi;
      const float tr = ur - vr, ti = ui - vi;
      re[i1] = tr * wr - ti * wi;
      im[i1] = tr * wi + ti * wr;
    }
    __syncthreads();
  }
}

// ---------------------------------------------------------------------------
// Kernel 1: per-channel Hann-windowed FFT magnitude, mean over batch.
// One block per channel; 16 sequential 4096-pt FFTs resident in LDS.
// Next batch's strided column is double-buffered into sStage with CDNA5
// async global->LDS loads, overlapped with the 12-stage FFT.
// ---------------------------------------------------------------------------
__global__ __launch_bounds__(NTHREADS) void k_mag(const float* __restrict__ x,
                                                  float* __restrict__ magT) {
  __shared__ float sRe[SEQ_LEN];
  __shared__ float sIm[SEQ_LEN];
  __shared__ float sStage[SEQ_LEN];
  __shared__ float sAcc[F_BINS];
  const int c = blockIdx.x;
  for (int f = threadIdx.x; f < F_BINS; f += blockDim.x) sAcc[f] = 0.0f;

  // prologue: async-stage batch 0's column
  for (int t = threadIdx.x; t < SEQ_LEN; t += blockDim.x)
    async_load_f32_to_lds(&x[(size_t)t * CHANNELS + c], &sStage[t]);
  wait_asynccnt0();
  __syncthreads();

  for (int b = 0; b < BATCH; ++b) {
    // stage -> bit-reversed positions with periodic Hann window
    for (int t = threadIdx.x; t < SEQ_LEN; t += blockDim.x) {
      const float h = 0.5f * (1.0f - __cosf(TWO_PI_F * (float)t / (float)SEQ_LEN));
      const float v = sStage[t] * h;
      const int p = brev12(t);
      sRe[p] = v;  sIm[p] = 0.0f;
    }
    __syncthreads();                     // all reads of sStage complete
    if (b + 1 < BATCH) {                 // overlap next column fetch with FFT
      for (int t = threadIdx.x; t < SEQ_LEN; t += blockDim.x)
        async_load_f32_to_lds(&x[((size_t)(b + 1) * SEQ_LEN + t) * CHANNELS + c],
                              &sStage[t]);
    }
    fft4096_dit(sRe, sIm, -1.0f);
    for (int f = threadIdx.x; f < F_BINS; f += blockDim.x) {
      const float rr = sRe[f], ii = sIm[f];
      sAcc[f] += sqrtf(rr * rr + ii * ii);
    }
    wait_asynccnt0();                    // staged column resident before reuse
    __syncthreads();
  }
  for (int f = threadIdx.x; f < F_BINS; f += blockDim.x)
    magT[(size_t)c * F_BINS + f] = sAcc[f] * (1.0f / (float)BATCH);
}

// ---------------------------------------------------------------------------
// Kernel 2: MLP layer 1 as fp32 WMMA GEMM: out1 = relu(magT(512x2049) @ w1(2049x64) + b1)
// V_WMMA_F32_16X16X4_F32: one 16x16 tile per wave, K stepped by 4.
// Main loop covers K in [0,2048) with no bounds checks (branch-free);
// a single register-masked tail WMMA handles k=2048.
// A 16x4 layout: lanes 0-15 hold M=lane, K={k0,k0+1}; lanes 16-31 K={k0+2,k0+3}.
// B 4x16 layout mirrors A with N striped across lanes.
// ---------------------------------------------------------------------------
__global__ __launch_bounds__(128) void k_gemm1(const float* __restrict__ magT,
                                               const float* __restrict__ w1,
                                               const float* __restrict__ b1,
                                               float* __restrict__ out1) {
  const int wave  = blockIdx.x * (blockDim.x >> 5) + (threadIdx.x >> 5);
  const int lane  = threadIdx.x & 31;
  const int half  = lane >> 4;        // 0 -> K pair {k0,k0+1}, 1 -> {k0+2,k0+3}
  const int r     = lane & 15;
  const int mTile = wave >> 2;        // 512/16 = 32 tiles in M
  const int nTile = wave & 3;         // 64/16  = 4 tiles in N
  const int m = mTile * 16 + r;
  const int n = nTile * 16 + r;
  const float* aRow = magT + (size_t)m * F_BINS;
  v8f acc = {};
  for (int k0 = 0; k0 < 2048; k0 += 4) {      // guard-free main loop
    const int ka = k0 + 2 * half;
    v2f a, b;
    a.x = aRow[ka];
    a.y = aRow[ka + 1];
    b.x = w1[(size_t)ka * 64 + n];
    b.y = w1[(size_t)(ka + 1) * 64 + n];
    acc = __builtin_amdgcn_wmma_f32_16x16x4_f32(
        /*neg_a=*/false, a, /*neg_b=*/false, b,
        /*c_mod=*/(short)0, acc, /*reuse_a=*/false, /*reuse_b=*/false);
  }
  // tail: K group {2048..2051}; only k=2048 is real. Loads are always
  // in-bounds; out-of-range lanes select 0 in registers (no EXEC churn).
  {
    const float aT = aRow[2048];
    const float bT = w1[(size_t)2048 * 64 + n];
    v2f a, b;
    a.x = half ? 0.0f : aT;  a.y = 0.0f;
    b.x = half ? 0.0f : bT;  b.y = 0.0f;
    acc = __builtin_amdgcn_wmma_f32_16x16x4_f32(
        false, a, false, b, (short)0, acc, false, false);
  }
  // C/D layout: VGPR j -> rows {j, j+8} split across lane halves, N = lane
  const float bias = b1[n];
  for (int j = 0; j < 8; ++j) {
    const int row = mTile * 16 + j + 8 * half;
    const float v = acc[j] + bias;
    out1[(size_t)row * 64 + n] = v > 0.0f ? v : 0.0f;
  }
}

// ---------------------------------------------------------------------------
// Kernel 3: MLP layers 2+3 + sigmoid + window computation (one thread/channel).
// ---------------------------------------------------------------------------
__global__ __launch_bounds__(NTHREADS) void k_mlp_tail(const float* __restrict__ out1,
                                                       const float* __restrict__ w2,
                                                       const float* __restrict__ b2,
                                                       const float* __restrict__ w3,
                                                       const float* __restrict__ b3,
                                                       int* __restrict__ winArr,
                                                       int* __restrict__ padArr) {
  const int c = blockIdx.x * blockDim.x + threadIdx.x;
  if (c >= CHANNELS) return;
  float h1[64];
  for (int k = 0; k < 64; ++k) h1[k] = out1[(size_t)c * 64 + k];
  float s = b3[0];
  for (int j = 0; j < 32; ++j) {
    float a = b2[j];
    for (int k = 0; k < 64; ++k) a += h1[k] * w2[k * 32 + j];
    a = a > 0.0f ? a : 0.0f;        // ReLU
    s += a * w3[j];
  }
  const float ratio = 1.0f / (1.0f + __expf(-s));       // sigmoid
  float wf = ratio * (float)SEQ_LEN;
  wf = fminf(fmaxf(wf, 10.0f), (float)(SEQ_LEN / 2));   // clip then trunc, like ref
  int win = (int)wf;
  if ((win & 1) == 0) win += 1;                         // force odd
  winArr[c] = win;
  padArr[c] = win >> 1;
}

// ---------------------------------------------------------------------------
// Kernel 4: adaptive moving average (trend) via reflect-pad + LDS prefix scan.
// One block per (b,c) column; padded column (6144 floats) lives in LDS.
// ---------------------------------------------------------------------------
__global__ __launch_bounds__(NTHREADS) void k_trend(const float* __restrict__ x,
                                                    const int* __restrict__ winArr,
                                                    const int* __restrict__ padArr,
                                                    float* __restrict__ trend) {
  __shared__ float sXp[XPLEN];
  __shared__ float sPart[NTHREADS];
  const int c = blockIdx.x & (CHANNELS - 1);
  const int b = blockIdx.x >> 9;                 // /512
  const float* xc = x + (size_t)b * SEQ_LEN * CHANNELS + c;
  // reflect padding (without edge): xp[-k] = x[k], xp[L-1+k] = x[L-1-k]
  for (int i = threadIdx.x; i < XPLEN; i += blockDim.x) {
    int src;
    if (i < PMAX)                 src = PMAX - i;
    else if (i < PMAX + SEQ_LEN)  src = i - PMAX;
    else                          src = 2 * SEQ_LEN + PMAX - 2 - i;
    sXp[i] = xc[(size_t)src * CHANNELS];
  }
  __syncthreads();
  // inclusive scan: per-thread chunk + serial scan of 256 partials
  const int base = threadIdx.x * CHUNK;
  float run = 0.0f;
  for (int k = 0; k < CHUNK; ++k) { run += sXp[base + k]; sXp[base + k] = run; }
  sPart[threadIdx.x] = run;
  __syncthreads();
  if (threadIdx.x == 0) {
    float acc = 0.0f;
    for (int i = 0; i < NTHREADS; ++i) { acc += sPart[i]; sPart[i] = acc; }
  }
  __syncthreads();
  const float off = (threadIdx.x > 0) ? sPart[threadIdx.x - 1] : 0.0f;
  for (int k = 0; k < CHUNK; ++k) sXp[base + k] += off;
  __syncthreads();
  const int win = winArr[c];
  const int pad = padArr[c];
  const float inv = 1.0f / (float)win;
  for (int t = threadIdx.x; t < SEQ_LEN; t += blockDim.x) {
    const int hi = t + PMAX + pad;               // inclusive-scan index
    const int lo = t + PMAX - pad;               // exclusive lower bound
    const float wsum = sXp[hi] - (lo > 0 ? sXp[lo - 1] : 0.0f);
    trend[((size_t)b * SEQ_LEN + t) * CHANNELS + c] = wsum * inv;
  }
}

// ---------------------------------------------------------------------------
// Kernel 5: seasonal = irfft( rfft(x - trend) * freq_attn ).
// Forward DIT (bitrev load -> natural spectrum), Hermitian-symmetric scaling,
// inverse DIF (natural -> bitrev output, un-reversed at the store).
// ---------------------------------------------------------------------------
__global__ __launch_bounds__(NTHREADS) void k_seasonal(const float* __restrict__ x,
                                                       const float* __restrict__ trend,
                                                       const float* __restrict__ attn,
                                                       float* __restrict__ seasonal) {
  __shared__ float sRe[SEQ_LEN];
  __shared__ float sIm[SEQ_LEN];
  const int c = blockIdx.x & (CHANNELS - 1);
  const int b = blockIdx.x >> 9;
  const size_t basebc = (size_t)b * SEQ_LEN * CHANNELS + c;
  for (int t = threadIdx.x; t < SEQ_LEN; t += blockDim.x) {
    const size_t idx = basebc + (size_t)t * CHANNELS;
    const float r = x[idx] - trend[idx];
    const int p = brev12(t);
    sRe[p] = r;  sIm[p] = 0.0f;
  }
  __syncthreads();
  fft4096_dit(sRe, sIm, -1.0f);
  // scale rFFT bins by attn[c][f]; keep Hermitian symmetry (attn is real)
  const float* ac = attn + (size_t)c * F_BINS;
  for (int f = threadIdx.x; f < F_BINS; f += blockDim.x) {
    const float g = ac[f];
    sRe[f] *= g;  sIm[f] *= g;
    if (f >= 1 && f < F_BINS - 1) {
      const int m = SEQ_LEN - f;
      sRe[m] *= g;  sIm[m] *= g;
    }
  }
  __syncthreads();
  fft4096_dif_inv(sRe, sIm);
  const float invN = 1.0f / (float)SEQ_LEN;
  for (int t = threadIdx.x; t < SEQ_LEN; t += blockDim.x)
    seasonal[basebc + (size_t)t * CHANNELS] = sRe[brev12(t)] * invN;
}

// ---------------------------------------------------------------------------
extern "C" void kernel_launch(void* const* d_in, const int* in_sizes, int n_in,
                              void* d_out, int out_size, void* d_ws, size_t ws_size,
                              hipStream_t stream) {
  const float* x         = (const float*)d_in[0];
  const float* freq_attn = (const float*)d_in[1];
  const float* w1        = (const float*)d_in[2];
  const float* b1        = (const float*)d_in[3];
  const float* w2        = (const float*)d_in[4];
  const float* b2        = (const float*)d_in[5];
  const float* w3        = (const float*)d_in[6];
  const float* b3        = (const float*)d_in[7];

  float* seasonal = (float*)d_out;
  float* trend    = seasonal + (size_t)BATCH * SEQ_LEN * CHANNELS;

  // workspace layout (~4.4 MB)
  float* magT  = (float*)d_ws;                                  // 512*2049 f32
  float* out1  = magT + (size_t)CHANNELS * F_BINS;              // 512*64   f32
  int*   winA  = (int*)(out1 + (size_t)CHANNELS * 64);          // 512 i32
  int*   padA  = winA + CHANNELS;                               // 512 i32

  k_mag     <<<CHANNELS,         NTHREADS, 0, stream>>>(x, magT);
  k_gemm1   <<<32,               128,      0, stream>>>(magT, w1, b1, out1);
  k_mlp_tail<<<2,                NTHREADS, 0, stream>>>(out1, w2, b2, w3, b3, winA, padA);
  k_trend   <<<BATCH * CHANNELS, NTHREADS, 0, stream>>>(x, winA, padA, trend);
  k_seasonal<<<BATCH * CHANNELS, NTHREADS, 0, stream>>>(x, trend, freq_attn, seasonal);
}